// BilinearDecoder_23751169146903
// MI455X (gfx1250) — compile-verified
//
#include <hip/hip_runtime.h>
#include <math.h>

// MI455X / gfx1250, wave32.
// Reference: logits[e,r] = u_e^T Q_r m_e ; out = log_softmax(logits, axis=-1)
// D = 64, R = 5 (checked implicitly via sizes in launcher).
//
// Primary path (needs ws >= N_MOVIES*320*4 = 128 MB):
//   stage1: T[m, r*64+k] = sum_l Q[r][k][l] * z_movie[m][l]
//           == GEMM  Z_movie(100000x64) x Qflat^T(64x320) via V_WMMA_F32_16X16X4_F32.
//           One wave per 16-row movie tile; A fragments held in registers and
//           reused across all 20 column tiles (20x less A gather traffic).
//   stage2: per edge gather u row + 5 t rows, 5 dot-64s + log_softmax (VALU,
//           L2-bandwidth bound). Output stores / index loads are non-temporal
//           so the 128MB T table + 51MB embedding tables stay L2-resident
//           (192MB L2; out(20MB)+indices(8MB) would otherwise overflow it).
// Fallback path: fused per-16-edge-tile WMMA kernel (no workspace needed).

typedef __attribute__((ext_vector_type(2))) float v2f;
typedef __attribute__((ext_vector_type(4))) float v4f;
typedef __attribute__((ext_vector_type(8))) float v8f;

// K-permutation used for both A and B fragments (sum over l is order-free):
//   l(ks=2*it+s, kk=2*h+c) = 8*it + 4*h + 2*s + c
// => A fragment (it): one float4 per lane at  row*64 + 8*it + 4*h
// => B fragment (it): one float4 per lane at  col*64 + 8*it + 4*h  (pairs for s=0,1)

// ---------------------------------------------------------------------------
// Stage 1: T = Z_movie x Qflat^T   (Qflat = Q viewed as [320][64], row c=r*64+k)
// One wave per 16-row movie tile; loops over 20 column tiles. T: [n_movies][320].
// ---------------------------------------------------------------------------
__global__ __launch_bounds__(256) void stage1_gemm(const float* __restrict__ zm,
                                                   const float* __restrict__ Qf,
                                                   float* __restrict__ T,
                                                   int n_movies) {
  const int lane = threadIdx.x & 31;
  const int wib  = threadIdx.x >> 5;
  const int sub  = lane & 15;   // row/col index within 16x16 tile
  const int h    = lane >> 4;   // half-wave
  const int i    = blockIdx.x * (blockDim.x >> 5) + wib;  // movie-row tile
  if (i >= (n_movies >> 4)) return;

  // A fragments for this row tile, kept live across all 20 column tiles.
  const float* arow = zm + (size_t)(i * 16 + sub) * 64 + 4 * h;
  v4f a4[8];
#pragma unroll
  for (int it = 0; it < 8; ++it) a4[it] = *(const v4f*)(arow + 8 * it);

  float* obase = T + (size_t)(i * 16 + 8 * h) * 320 + sub;

  for (int j = 0; j < 20; ++j) {               // column tile over 320 = R*64
    const float* brow = Qf + (size_t)(j * 16 + sub) * 64 + 4 * h;
    v8f acc = {};
#pragma unroll
    for (int it = 0; it < 8; ++it) {
      v4f b4 = *(const v4f*)(brow + 8 * it);
      v2f aA = {a4[it].x, a4[it].y};
      v2f aB = {a4[it].z, a4[it].w};
      v2f b0 = {b4.x, b4.y};
      v2f b1 = {b4.z, b4.w};
      acc = __builtin_amdgcn_wmma_f32_16x16x4_f32(false, aA, false, b0, (short)0, acc, false, false);
      acc = __builtin_amdgcn_wmma_f32_16x16x4_f32(false, aB, false, b1, (short)0, acc, false, false);
    }
    // D layout: VGPR v holds rows v (lanes 0-15) and v+8 (lanes 16-31), col = sub.
    float* orow = obase + j * 16;
#pragma unroll
    for (int v = 0; v < 8; ++v) orow[(size_t)v * 320] = acc[v];
  }
}

// ---------------------------------------------------------------------------
// Stage 2: per-edge dots + log_softmax. 8 lanes per edge, 4 edges per wave.
// ---------------------------------------------------------------------------
__global__ __launch_bounds__(256) void stage2_edges(const float* __restrict__ zu,
                                                    const float* __restrict__ T,
                                                    const int* __restrict__ eidx,
                                                    float* __restrict__ out, int E) {
  const int lane = threadIdx.x & 31;
  const int q = lane & 7;     // lane within edge group (feature slice)
  const int g = lane >> 3;    // edge group within wave (0..3)
  const int waveId = (blockIdx.x * blockDim.x + threadIdx.x) >> 5;
  const int e = waveId * 4 + g;
  if (e >= E) return;                       // uniform per 8-lane group

  const int urow = __builtin_nontemporal_load(eidx + e);      // user index
  const int mrow = __builtin_nontemporal_load(eidx + E + e);  // movie index
  const float* up = zu + (size_t)urow * 64 + q * 8;
  const float* tp = T + (size_t)mrow * 320 + q * 8;
  v4f u0 = *(const v4f*)(up);
  v4f u1 = *(const v4f*)(up + 4);

  float lg[5];
#pragma unroll
  for (int r = 0; r < 5; ++r) {
    v4f t0 = *(const v4f*)(tp + 64 * r);
    v4f t1 = *(const v4f*)(tp + 64 * r + 4);
    float p = u0.x * t0.x + u0.y * t0.y + u0.z * t0.z + u0.w * t0.w +
              u1.x * t1.x + u1.y * t1.y + u1.z * t1.z + u1.w * t1.w;
    p += __shfl_xor(p, 1, 32);
    p += __shfl_xor(p, 2, 32);
    p += __shfl_xor(p, 4, 32);              // all 8 lanes of group hold logit r
    lg[r] = p;
  }
  float mx = fmaxf(fmaxf(fmaxf(lg[0], lg[1]), fmaxf(lg[2], lg[3])), lg[4]);
  float s = expf(lg[0] - mx) + expf(lg[1] - mx) + expf(lg[2] - mx) +
            expf(lg[3] - mx) + expf(lg[4] - mx);
  float lse = mx + logf(s);
  float sel = lg[0];
  sel = (q == 1) ? lg[1] : sel;
  sel = (q == 2) ? lg[2] : sel;
  sel = (q == 3) ? lg[3] : sel;
  sel = (q == 4) ? lg[4] : sel;
  if (q < 5) __builtin_nontemporal_store(sel - lse, out + (size_t)e * 5 + q);
}

// ---------------------------------------------------------------------------
// Fallback: fused WMMA kernel, one wave per 16-edge tile (no workspace).
//   T_rj = M_tile(16x64) x Q_r^T tile, then logits = rowsum(T_rj * U_j).
// ---------------------------------------------------------------------------
__global__ __launch_bounds__(256) void fused_wmma(const float* __restrict__ zu,
                                                  const float* __restrict__ zm,
                                                  const int* __restrict__ eidx,
                                                  const float* __restrict__ Qf,
                                                  float* __restrict__ out, int E) {
  __shared__ float smem[8][80];   // [wave][edge*5 + r]
  const int lane = threadIdx.x & 31;
  const int wib  = threadIdx.x >> 5;
  const int sub  = lane & 15;
  const int h    = lane >> 4;
  const int tile = blockIdx.x * (blockDim.x >> 5) + wib;
  const int nTiles = (E + 15) >> 4;
  if (tile >= nTiles) return;
  const int eb = tile * 16;

  // A = movie rows for the 16 edges of this tile.
  int ea = eb + sub; if (ea >= E) ea = E - 1;
  const float* arow = zm + (size_t)eidx[E + ea] * 64 + 4 * h;
  v4f a4[8];
#pragma unroll
  for (int it = 0; it < 8; ++it) a4[it] = *(const v4f*)(arow + 8 * it);

  // U tiles in C/D layout: U[j][v] (this lane) = zu[user(edge v+8h)][16*j + sub]
  v8f U[4];
#pragma unroll
  for (int v = 0; v < 8; ++v) {
    int ev = eb + v + 8 * h; if (ev >= E) ev = E - 1;
    const float* up = zu + (size_t)eidx[ev] * 64 + sub;
#pragma unroll
    for (int j = 0; j < 4; ++j) U[j][v] = up[16 * j];
  }

#pragma unroll
  for (int r = 0; r < 5; ++r) {
    v8f partial = {};
#pragma unroll
    for (int j = 0; j < 4; ++j) {
      // B[kk][n] = Q[r][16j+n][l]  ==  Qflat row (r*4+j)*16 + sub
      const float* brow = Qf + (size_t)((r * 4 + j) * 16 + sub) * 64 + 4 * h;
      v8f acc = {};
#pragma unroll
      for (int it = 0; it < 8; ++it) {
        v4f b4 = *(const v4f*)(brow + 8 * it);
        v2f aA = {a4[it].x, a4[it].y};
        v2f aB = {a4[it].z, a4[it].w};
        v2f b0 = {b4.x, b4.y};
        v2f b1 = {b4.z, b4.w};
        acc = __builtin_amdgcn_wmma_f32_16x16x4_f32(false, aA, false, b0, (short)0, acc, false, false);
        acc = __builtin_amdgcn_wmma_f32_16x16x4_f32(false, aB, false, b1, (short)0, acc, false, false);
      }
      partial += acc * U[j];          // elementwise; same C/D layout
    }
    // logit[v+8h] = sum over the 16 lanes of this half of partial[v]
#pragma unroll
    for (int v = 0; v < 8; ++v) {
      float p = partial[v];
      p += __shfl_xor(p, 1, 32);
      p += __shfl_xor(p, 2, 32);
      p += __shfl_xor(p, 4, 32);
      p += __shfl_xor(p, 8, 32);
      if (sub == v) smem[wib][(v + 8 * h) * 5 + r] = p;
    }
  }

  // Within-wave LDS RAW: DS pipe is in-order per wave; make it explicit.
  asm volatile("s_wait_dscnt 0" ::: "memory");

  if (lane < 16) {
    const int e = eb + lane;
    if (e < E) {
      float l0 = smem[wib][lane * 5 + 0];
      float l1 = smem[wib][lane * 5 + 1];
      float l2 = smem[wib][lane * 5 + 2];
      float l3 = smem[wib][lane * 5 + 3];
      float l4 = smem[wib][lane * 5 + 4];
      float mx = fmaxf(fmaxf(fmaxf(l0, l1), fmaxf(l2, l3)), l4);
      float s = expf(l0 - mx) + expf(l1 - mx) + expf(l2 - mx) +
                expf(l3 - mx) + expf(l4 - mx);
      float lse = mx + logf(s);
      float* op = out + (size_t)e * 5;
      __builtin_nontemporal_store(l0 - lse, op + 0);
      __builtin_nontemporal_store(l1 - lse, op + 1);
      __builtin_nontemporal_store(l2 - lse, op + 2);
      __builtin_nontemporal_store(l3 - lse, op + 3);
      __builtin_nontemporal_store(l4 - lse, op + 4);
    }
  }
}

// ---------------------------------------------------------------------------
extern "C" void kernel_launch(void* const* d_in, const int* in_sizes, int n_in,
                              void* d_out, int out_size, void* d_ws, size_t ws_size,
                              hipStream_t stream) {
  const float* zu  = (const float*)d_in[0];   // [N_USERS][64]
  const float* zm  = (const float*)d_in[1];   // [N_MOVIES][64]
  const int* eidx  = (const int*)d_in[2];     // [2][E]
  const float* Qf  = (const float*)d_in[3];   // [5][64][64] == [320][64]
  float* out = (float*)d_out;                 // [E][5]

  const int n_movies = in_sizes[1] / 64;
  const int E = in_sizes[2] / 2;
  const size_t tbytes = (size_t)n_movies * 320 * sizeof(float);

  if (ws_size >= tbytes && (n_movies % 16) == 0) {
    float* T = (float*)d_ws;
    const int rowTiles = n_movies / 16;              // one wave per row tile
    const int blocks1  = (rowTiles + 7) / 8;         // 8 waves per block
    stage1_gemm<<<blocks1, 256, 0, stream>>>(zm, Qf, T, n_movies);
    const int blocks2 = (E + 31) / 32;               // 32 edges per block
    stage2_edges<<<blocks2, 256, 0, stream>>>(zu, T, eidx, out, E);
  } else {
    const int tiles  = (E + 15) / 16;
    const int blocks = (tiles + 7) / 8;
    fused_wmma<<<blocks, 256, 0, stream>>>(zu, zm, eidx, Qf, out, E);
  }
}